// SparseArch_6416681140553
// MI455X (gfx1250) — compile-verified
//
#include <hip/hip_runtime.h>
#include <stdint.h>

typedef __attribute__((ext_vector_type(2))) float v2f;
typedef __attribute__((ext_vector_type(8))) float v8f;

#define NFEAT 4
#define DDIM 64
#define MAXROWS 16
#define WAVES_PER_BLOCK 8
#define TPB (WAVES_PER_BLOCK * 32)

struct PoolArgs {
  const int*   idx[NFEAT];
  const int*   len[NFEAT];
  const float* emb[NFEAT];
  const float* pw[NFEAT];
  int          L[NFEAT];
  int          B;
  float*       pred;     // [B, NFEAT*DDIM], row-major
  float*       partial;  // one float per block
};

// One wave per (feature, bag). Async-gather up to L rows (256B each) of the
// embedding table into LDS (2 rows per b128 issue, ASYNCcnt-tracked), then
// position-weighted accumulate: lane owns dims {2*lane, 2*lane+1}.
__global__ __launch_bounds__(TPB) void pool_kernel(PoolArgs a) {
  __shared__ float tile[WAVES_PER_BLOCK][MAXROWS * DDIM];   // 32 KB
  __shared__ float blockpart[WAVES_PER_BLOCK];

  const int lane = threadIdx.x & 31;
  const int wv   = threadIdx.x >> 5;
  const int w    = blockIdx.x * WAVES_PER_BLOCK + wv;       // global wave id
  const bool valid = (w < NFEAT * a.B);

  const int feat = valid ? (w / a.B) : 0;
  const int bag  = valid ? (w - feat * a.B) : 0;

  int L = a.L[feat];
  if (L > MAXROWS) L = MAXROWS;
  const int   len = valid ? a.len[feat][bag] : 0;
  const int*  idx = a.idx[feat] + (size_t)bag * a.L[feat];
  const float* emb = a.emb[feat];
  const float* pw  = a.pw[feat];

  // ---- async gather: lanes 0-15 fetch row l0, lanes 16-31 fetch row l0+1 ----
  const int sub = lane >> 4;        // which of the two rows
  const int c16 = lane & 15;        // 16-byte chunk within the 256B row
  for (int l0 = 0; l0 < L; l0 += 2) {
    int row = l0 + sub;
    if (row >= L) row = L - 1;      // redundant (weight-masked) load, harmless
    const int rid = idx[row];
    uint64_t gaddr = (uint64_t)(uintptr_t)(emb + (size_t)rid * DDIM)
                   + (uint64_t)(c16 * 16);
    uint32_t laddr = (uint32_t)(uintptr_t)&tile[wv][row * DDIM + c16 * 4];
    asm volatile("global_load_async_to_lds_b128 %0, %1, off"
                 :: "v"(laddr), "v"(gaddr) : "memory");
  }
  asm volatile("s_wait_asynccnt 0" ::: "memory");

  // ---- weighted accumulation ----
  float ax = 0.f, ay = 0.f;
  for (int l = 0; l < L; ++l) {
    const float wgt = (l < len) ? pw[l] : 0.f;
    ax = fmaf(wgt, tile[wv][l * DDIM + 2 * lane],     ax);
    ay = fmaf(wgt, tile[wv][l * DDIM + 2 * lane + 1], ay);
  }

  if (valid) {
    float* o = a.pred + (size_t)bag * (NFEAT * DDIM) + feat * DDIM + 2 * lane;
    o[0] = ax;   // d_out+1 is only 4B aligned -> scalar stores
    o[1] = ay;
  }

  // ---- deterministic partial sum for the loss ----
  float v = valid ? (ax + ay) : 0.f;
  for (int off = 16; off > 0; off >>= 1) v += __shfl_down(v, off, 32);
  if (lane == 0) blockpart[wv] = v;
  __syncthreads();
  if (threadIdx.x == 0) {
    float s = 0.f;
    for (int i = 0; i < WAVES_PER_BLOCK; ++i) s += blockpart[i];
    a.partial[blockIdx.x] = s;
  }
}

// Single-block final reduction. 256 threads fold `np` partials into 256 LDS
// values, then a V_WMMA_F32_16X16X4_F32 with A=ones reduces 64 values per
// issue (C-accumulated over 4 tiles), finished with a wave shuffle reduce.
__global__ __launch_bounds__(256) void loss_kernel(const float* __restrict__ partial,
                                                   int np, float* loss_out,
                                                   float inv_n) {
  __shared__ float s[256];
  float v = 0.f;
  for (int i = threadIdx.x; i < np; i += 256) v += partial[i];  // fixed order
  s[threadIdx.x] = v;
  __syncthreads();

  const int lane = threadIdx.x & 31;
  float r;
#if __has_builtin(__builtin_amdgcn_wmma_f32_16x16x4_f32)
  // Executed uniformly by every wave (EXEC must be all 1s for WMMA).
  v2f ones = {1.f, 1.f};
  v8f c = {};
  for (int t = 0; t < 4; ++t) {
    // Bijective tile mapping of s[t*64 .. t*64+63] onto B[4x16]; exact
    // placement is irrelevant since D[m,n] = sum_k B[k,n] and we sum all n.
    int base = t * 64 + (lane >> 4) * 32 + (lane & 15);
    v2f b = {s[base], s[base + 16]};
    c = __builtin_amdgcn_wmma_f32_16x16x4_f32(
        /*neg_a=*/false, ones, /*neg_b=*/false, b,
        /*c_mod=*/(short)0, c, /*reuse_a=*/false, /*reuse_b=*/false);
  }
  // D row M=0 lives in c[0] lanes 0-15 (lanes 16-31 hold identical row M=8).
  r = (lane < 16) ? c[0] : 0.f;
#else
  r = 0.f;
  for (int i = lane; i < 256; i += 32) r += s[i];
#endif
  for (int off = 16; off > 0; off >>= 1) r += __shfl_down(r, off, 32);
  if (threadIdx.x == 0) loss_out[0] = r * inv_n;
}

extern "C" void kernel_launch(void* const* d_in, const int* in_sizes, int n_in,
                              void* d_out, int out_size, void* d_ws, size_t ws_size,
                              hipStream_t stream) {
  (void)n_in; (void)out_size; (void)ws_size;
  PoolArgs a;
  const int B = in_sizes[1];  // lengths_0 element count
  for (int f = 0; f < NFEAT; ++f) {
    a.idx[f] = (const int*)  d_in[4 * f + 0];
    a.len[f] = (const int*)  d_in[4 * f + 1];
    a.emb[f] = (const float*)d_in[4 * f + 2];
    a.pw[f]  = (const float*)d_in[4 * f + 3];
    a.L[f]   = in_sizes[4 * f + 3];           // pw length == max feature length
  }
  a.B = B;
  a.pred    = (float*)d_out + 1;              // d_out[0] = loss, then pred
  a.partial = (float*)d_ws;

  const int nwaves  = NFEAT * B;
  const int nblocks = (nwaves + WAVES_PER_BLOCK - 1) / WAVES_PER_BLOCK;
  pool_kernel<<<nblocks, TPB, 0, stream>>>(a);

  const float inv_n = 1.0f / ((float)B * (float)(NFEAT * DDIM));
  loss_kernel<<<1, 256, 0, stream>>>((const float*)d_ws, nblocks,
                                     (float*)d_out, inv_n);
}